// EdgeEncoding_72816875537094
// MI455X (gfx1250) — compile-verified
//
#include <hip/hip_runtime.h>

typedef __attribute__((ext_vector_type(2))) float v2f;
typedef __attribute__((ext_vector_type(2))) int   v2i;
typedef __attribute__((ext_vector_type(4))) float v4f;
typedef __attribute__((ext_vector_type(8))) float v8f;

#define NB 64    // graphs
#define NN 64    // nodes per graph
#define NE 128   // edges per graph
#define ED 64    // edge feature dim

// ---------------------------------------------------------------------------
// Kernel 1: scores[g] = edge_attr[g,:] . W + bias   for g in [0, NB*NE)
// 2 MiB read, trivially bandwidth bound. W/bias loads are uniform -> scalar.
// ---------------------------------------------------------------------------
__global__ void edge_scores_kernel(const float* __restrict__ edge_attr,
                                   const float* __restrict__ W,
                                   const float* __restrict__ bias,
                                   float* __restrict__ scores) {
    int g = blockIdx.x * blockDim.x + threadIdx.x;   // 0..8191
    const v4f* a4 = (const v4f*)(edge_attr + (size_t)g * ED);
    const v4f* w4 = (const v4f*)W;
    float s = 0.f;
#pragma unroll
    for (int i = 0; i < ED / 4; ++i) {
        v4f a = a4[i];
        v4f w = w4[i];
        s += a.x * w.x + a.y * w.y + a.z * w.z + a.w * w.w;
    }
    scores[g] = s + bias[0];
}

// ---------------------------------------------------------------------------
// Kernel 2: one wave32 per 16-row output tile.
//   D(16x16) += A(16x4 masks, f32) x B(4x16, col0=scores col1=ones)
//   K-loop of 32 over E=128 with V_WMMA_F32_16X16X4_F32 (exact f32).
// A layout (32-bit 16x4): lane l -> row M=l%16, K = 2*(l>=16)+{0,1}  (2 VGPRs)
// B layout (4x16 32-bit): VGPR v -> row K = v + 2*(l>=16), col N = l%16
// D layout (16x16 32-bit): VGPR v -> row M = v + 8*(l>=16), col N = l%16
// ---------------------------------------------------------------------------
__global__ void path_score_kernel(const int*   __restrict__ edge_paths,
                                  const float* __restrict__ scores,
                                  float*       __restrict__ out) {
    const int lane = threadIdx.x & 31;
    const int wave = threadIdx.x >> 5;           // 8 waves per block
    const int half = lane >> 4;                  // 0: K pair {0,1}, 1: {2,3}
    const int n    = lane & 15;                  // column / row-in-tile index
    const int tile = blockIdx.x * 8 + wave;      // 0..16383
    const int row0 = tile * 16;                  // global flat output row
    const int b    = row0 >> 12;                 // row0 / (N*N)

    const int*   arow = edge_paths + (size_t)(row0 + n) * NE;  // my A row
    const float* sb   = scores + b * NE;
    const int    koff = 2 * half;

    v8f c0 = {};
    v8f c1 = {};
#pragma unroll
    for (int e0 = 0; e0 < NE; e0 += 8) {
        // ---- chunk 0: K = e0 .. e0+3 ----
        v2i m0 = *(const v2i*)(arow + e0 + koff);          // 8B, exact 0/1
        v2f a0; a0.x = (float)m0.x; a0.y = (float)m0.y;
        v2f s0 = *(const v2f*)(sb + e0 + koff);
        v2f b0;
        b0.x = (n == 0) ? s0.x : ((n == 1) ? 1.0f : 0.0f);
        b0.y = (n == 0) ? s0.y : ((n == 1) ? 1.0f : 0.0f);
        c0 = __builtin_amdgcn_wmma_f32_16x16x4_f32(
                 false, a0, false, b0, (short)0, c0, false, false);

        // ---- chunk 1: K = e0+4 .. e0+7 (independent accumulator) ----
        v2i m1 = *(const v2i*)(arow + e0 + 4 + koff);
        v2f a1; a1.x = (float)m1.x; a1.y = (float)m1.y;
        v2f s1 = *(const v2f*)(sb + e0 + 4 + koff);
        v2f b1;
        b1.x = (n == 0) ? s1.x : ((n == 1) ? 1.0f : 0.0f);
        b1.y = (n == 0) ? s1.y : ((n == 1) ? 1.0f : 0.0f);
        c1 = __builtin_amdgcn_wmma_f32_16x16x4_f32(
                 false, a1, false, b1, (short)0, c1, false, false);
    }
    v8f c = c0 + c1;

    // D[M,0] = num, D[M,1] = den. Pair them with a lane0<->lane1 shuffle,
    // lanes with N==0 (lanes 0 and 16) store rows v and v+8 respectively.
#pragma unroll
    for (int v = 0; v < 8; ++v) {
        float num = c[v];
        float den = __shfl_xor(c[v], 1, 32);   // lane^1: N==1 holds den
        if (n == 0) {
            out[row0 + 8 * half + v] = num / (den + 1e-8f);
        }
    }
}

// ---------------------------------------------------------------------------
// Inputs (setup_inputs order): edge_attr [8192*64] f32, W [64] f32, b [1] f32,
// edge_paths [64*64*64*128] i32, ptr [65] i64 (unused: uniform graphs).
// Output: [64*64*64] f32. Workspace: 32 KiB for scores.
// ---------------------------------------------------------------------------
extern "C" void kernel_launch(void* const* d_in, const int* in_sizes, int n_in,
                              void* d_out, int out_size, void* d_ws, size_t ws_size,
                              hipStream_t stream) {
    const float* edge_attr  = (const float*)d_in[0];
    const float* W          = (const float*)d_in[1];
    const float* bias       = (const float*)d_in[2];
    const int*   edge_paths = (const int*)d_in[3];
    (void)in_sizes; (void)n_in; (void)out_size; (void)ws_size;

    float* scores = (float*)d_ws;          // NB*NE = 8192 floats
    float* out    = (float*)d_out;         // NB*NN*NN floats

    // 8192 scores, 256 threads/block
    edge_scores_kernel<<<NB * NE / 256, 256, 0, stream>>>(edge_attr, W, bias, scores);

    // 16384 tiles of 16 rows, 8 wave32 tiles per 256-thread block
    const int tiles = NB * NN * NN / 16;   // 16384
    path_score_kernel<<<tiles / 8, 256, 0, stream>>>(edge_paths, scores, out);
}